// GCN_EdgePool_81973745811866
// MI455X (gfx1250) — compile-verified
//
#include <hip/hip_runtime.h>
#include <hip/hip_bf16.h>

// ---------------- problem constants ----------------
constexpr int N      = 50000;
constexpr int E      = 800000;
constexpr int B      = 64;
constexpr int IN_DIM = 128;
constexpr int HID    = 64;
constexpr int SORT_N = 1 << 20;   // next pow2 >= E

typedef __attribute__((ext_vector_type(2))) float v2f;
typedef __attribute__((ext_vector_type(8))) float v8f;

// ---------------- helpers ----------------
__device__ __forceinline__ unsigned floatFlip(float f) {
    unsigned u = __float_as_uint(f);
    unsigned mask = (u >> 31) ? 0xFFFFFFFFu : 0x80000000u;
    return u ^ mask;   // monotonic unsigned key
}
__device__ __forceinline__ float floatUnflip(unsigned u) {
    unsigned mask = (u >> 31) ? 0x80000000u : 0xFFFFFFFFu;
    return __uint_as_float(u ^ mask);
}
#define FLIP_NEG_INF 0x007FFFFFu   // floatFlip(-inf)

// ---------------- kernels ----------------

// deg=1 (self loop), remaining=1, cluster=0, sc=1, segmax=-inf, den=0, new_batch=0
__global__ void k_init_nodes(float* deg, int* remaining, int* cluster, float* sc,
                             unsigned* mseg, float* den, int* nbatch) {
    int i = blockIdx.x * blockDim.x + threadIdx.x;
    if (i >= N) return;
    deg[i] = 1.0f; remaining[i] = 1; cluster[i] = 0; sc[i] = 1.0f;
    mseg[i] = FLIP_NEG_INF; den[i] = 0.0f; nbatch[i] = 0;
}

__global__ void k_deg_edges(const int* col, float* deg) {
    int e = blockIdx.x * blockDim.x + threadIdx.x;
    if (e >= E) return;
    atomicAdd(&deg[col[e]], 1.0f);
}

__global__ void k_dinv(float* deg) {
    int i = blockIdx.x * blockDim.x + threadIdx.x;
    if (i >= N) return;
    float d = deg[i];                   // >= 1 always (self loop)
    deg[i] = rsqrtf(d);
}

// h = x @ W  via V_WMMA_F32_16X16X4_F32.  grid = N/16 blocks of 128 threads
// (4 wave32 per block, wave w owns output columns [16w,16w+16)).
__global__ void k_gemm_wmma(const float* __restrict__ x, const float* __restrict__ W,
                            float* __restrict__ h) {
    const int wave = threadIdx.x >> 5;
    const int lane = threadIdx.x & 31;
    const int row0 = blockIdx.x * 16;        // N % 16 == 0  -> no tail
    const int col0 = wave * 16;
    const int m    = lane & 15;              // A-matrix row  (lanes 0-15 / 16-31)
    const int nn   = lane & 15;              // B/C/D column
    const int hi   = lane >> 4;              // 0: K pair {0,1}, 1: K pair {2,3}
    const int koff = hi * 2;

    const float* __restrict__ xrow = x + (size_t)(row0 + m) * IN_DIM;

    v8f acc = {};
    #pragma unroll
    for (int kk = 0; kk < IN_DIM / 4; ++kk) {
        const int kb = kk * 4 + koff;
        v2f a, bb;
        a.x  = xrow[kb];
        a.y  = xrow[kb + 1];
        bb.x = W[(size_t)kb * HID + col0 + nn];
        bb.y = W[(size_t)(kb + 1) * HID + col0 + nn];
        acc = __builtin_amdgcn_wmma_f32_16x16x4_f32(
                  /*neg_a=*/false, a, /*neg_b=*/false, bb,
                  /*c_mod=*/(short)0, acc, /*reuse_a=*/false, /*reuse_b=*/false);
    }
    #pragma unroll
    for (int i = 0; i < 8; ++i) {
        const int mm = i + hi * 8;
        h[(size_t)(row0 + mm) * HID + col0 + nn] = acc[i];
    }
}

// hagg[i][j] = h[i][j] * dinv[i]^2   (self-loop term)
__global__ void k_selfloop(const float* h, const float* dinv, float* hagg) {
    long long idx = (long long)blockIdx.x * blockDim.x + threadIdx.x;
    if (idx >= (long long)N * HID) return;
    int i = (int)(idx >> 6);
    float di = dinv[i];
    hagg[idx] = h[idx] * di * di;
}

// hagg[dst][j] += h[src][j] * dinv[src]*dinv[dst]   per (edge, col)
__global__ void k_edge_scatter(const float* __restrict__ h, const float* __restrict__ dinv,
                               const int* __restrict__ src, const int* __restrict__ dst,
                               float* __restrict__ hagg) {
    long long idx = (long long)blockIdx.x * blockDim.x + threadIdx.x;
    if (idx >= (long long)E * HID) return;
    int e = (int)(idx >> 6), j = (int)(idx & 63);
    int s = src[e], d = dst[e];
    float w = dinv[s] * dinv[d];
    atomicAdd(&hagg[(size_t)d * HID + j], h[(size_t)s * HID + j] * w);
}

// hrelu = relu(hagg + b)   (in place)
__global__ void k_bias_relu(float* hagg, const float* bias) {
    long long idx = (long long)blockIdx.x * blockDim.x + threadIdx.x;
    if (idx >= (long long)N * HID) return;
    int j = (int)(idx & 63);
    hagg[idx] = fmaxf(hagg[idx] + bias[j], 0.0f);
}

// e-score per edge + segment-max over dst (fused)
__global__ void k_edge_score(const float* __restrict__ hrelu,
                             const int* __restrict__ src, const int* __restrict__ dst,
                             const float* __restrict__ w_pool, const float* __restrict__ b_pool,
                             float* escore, unsigned* mseg) {
    int e = blockIdx.x * blockDim.x + threadIdx.x;
    if (e >= E) return;
    int s = src[e], d = dst[e];
    const float* hs = hrelu + (size_t)s * HID;
    const float* hd = hrelu + (size_t)d * HID;
    float acc = b_pool[0];
    #pragma unroll 8
    for (int j = 0; j < HID; ++j)
        acc += hs[j] * w_pool[j] + hd[j] * w_pool[HID + j];
    escore[e] = acc;
    atomicMax(&mseg[d], floatFlip(acc));
}

// ex = exp(e - m[dst]); den[dst] += ex   (escore overwritten with ex)
__global__ void k_ex_den(float* escore, const int* dst, const unsigned* mseg, float* den) {
    int e = blockIdx.x * blockDim.x + threadIdx.x;
    if (e >= E) return;
    int d = dst[e];
    float ex = __expf(escore[e] - floatUnflip(mseg[d]));
    escore[e] = ex;
    atomicAdd(&den[d], ex);
}

// score = ex/den[dst] + 0.5
__global__ void k_score(float* escore, const int* dst, const float* den) {
    int e = blockIdx.x * blockDim.x + threadIdx.x;
    if (e >= E) return;
    escore[e] = escore[e] / den[dst[e]] + 0.5f;
}

// pack keys for a stable ascending sort of (-score, idx)
__global__ void k_init_keys(const float* score, unsigned long long* keys) {
    int i = blockIdx.x * blockDim.x + threadIdx.x;
    if (i >= SORT_N) return;
    if (i < E) {
        unsigned hi = floatFlip(-score[i]);
        keys[i] = ((unsigned long long)hi << 32) | (unsigned)i;
    } else {
        keys[i] = ~0ull;   // padding sorts to the end
    }
}

__global__ void k_bitonic_step(unsigned long long* keys, int j, int k) {
    int i = blockIdx.x * blockDim.x + threadIdx.x;
    int ixj = i ^ j;
    if (ixj <= i) return;
    unsigned long long a = keys[i], b2 = keys[ixj];
    bool asc = ((i & k) == 0);
    if (asc ? (a > b2) : (a < b2)) { keys[i] = b2; keys[ixj] = a; }
}

// sequential greedy matching (reference lax.scan semantics) + remaining ranks
__global__ void k_greedy(const unsigned long long* keys,
                         const int* src, const int* dst,
                         int* remaining, int* cluster, float* scbuf,
                         const float* score, int* counters) {
    if (threadIdx.x != 0 || blockIdx.x != 0) return;
    int nm = 0;
    for (int i = 0; i < SORT_N; ++i) {
        unsigned idx = (unsigned)(keys[i] & 0xFFFFFFFFull);
        if (idx >= (unsigned)E) break;          // padding: sorted to end
        int s = src[idx], d = dst[idx];
        if (remaining[s] && remaining[d]) {
            remaining[s] = 0; remaining[d] = 0;
            cluster[s] = nm; cluster[d] = nm;
            scbuf[nm]  = score[idx];
            ++nm;
        }
    }
    int rr = 0;
    for (int i = 0; i < N; ++i)
        if (remaining[i]) { cluster[i] = nm + rr; ++rr; }
    counters[0] = nm + rr;   // total clusters
}

__global__ void k_clear_newx(float* newx) {
    long long idx = (long long)blockIdx.x * blockDim.x + threadIdx.x;
    if (idx >= (long long)N * HID) return;
    newx[idx] = 0.0f;
}

// new_x[cluster[i]] += hrelu[i]; new_batch[cluster[i]] = max(batch[i])
__global__ void k_pool_scatter(const float* __restrict__ hrelu, const int* __restrict__ cluster,
                               const int* __restrict__ batch, float* newx, int* nbatch) {
    long long idx = (long long)blockIdx.x * blockDim.x + threadIdx.x;
    if (idx >= (long long)N * HID) return;
    int i = (int)(idx >> 6), j = (int)(idx & 63);
    int c = cluster[i];
    atomicAdd(&newx[(size_t)c * HID + j], hrelu[idx]);
    if (j == 0) atomicMax(&nbatch[c], batch[i]);
}

__global__ void k_scale_newx(float* newx, const float* scbuf) {
    long long idx = (long long)blockIdx.x * blockDim.x + threadIdx.x;
    if (idx >= (long long)N * HID) return;
    newx[idx] *= scbuf[(int)(idx >> 6)];
}

__global__ void k_clear_readout(float* meanb, unsigned* mxb, float* cntb) {
    int i = blockIdx.x * blockDim.x + threadIdx.x;
    if (i < B * HID) { meanb[i] = 0.0f; mxb[i] = FLIP_NEG_INF; }
    if (i < B) cntb[i] = 0.0f;
}

__global__ void k_readout_scatter(const float* __restrict__ newx, const int* __restrict__ nbatch,
                                  const int* __restrict__ counters,
                                  float* meanb, unsigned* mxb, float* cntb) {
    long long idx = (long long)blockIdx.x * blockDim.x + threadIdx.x;
    if (idx >= (long long)N * HID) return;
    int c = (int)(idx >> 6), j = (int)(idx & 63);
    if (c >= counters[0]) return;               // valid clusters only
    int bb = nbatch[c];
    float v = newx[(size_t)c * HID + j];
    atomicAdd(&meanb[(size_t)bb * HID + j], v);
    atomicMax(&mxb[(size_t)bb * HID + j], floatFlip(v));
    if (j == 0) atomicAdd(&cntb[bb], 1.0f);
}

__global__ void k_finalize(const float* meanb, const unsigned* mxb, const float* cntb,
                           float* out) {
    int idx = blockIdx.x * blockDim.x + threadIdx.x;
    if (idx >= B * HID) return;
    int bb = idx >> 6, j = idx & 63;
    float cnt  = fmaxf(cntb[bb], 1.0f);
    float mean = meanb[idx] / cnt;
    float mx   = floatUnflip(mxb[idx]);
    if (!isfinite(mx)) mx = 0.0f;
    out[(size_t)bb * (2 * HID) + j]       = mean;
    out[(size_t)bb * (2 * HID) + HID + j] = mx;
}

// ---------------- host ----------------
static inline size_t alignUp(size_t v, size_t a) { return (v + a - 1) / a * a; }

extern "C" void kernel_launch(void* const* d_in, const int* in_sizes, int n_in,
                              void* d_out, int out_size, void* d_ws, size_t ws_size,
                              hipStream_t stream) {
    (void)in_sizes; (void)n_in; (void)out_size; (void)ws_size;

    const float* x      = (const float*)d_in[0];                 // N x 128
    const int*   eidx   = (const int*)  d_in[1];                 // 2 x E
    const int*   batch  = (const int*)  d_in[2];                 // N
    const float* W      = (const float*)d_in[3];                 // 128 x 64
    const float* bias   = (const float*)d_in[4];                 // 64
    const float* w_pool = (const float*)d_in[5];                 // 128
    const float* b_pool = (const float*)d_in[6];                 // 1
    float*       out    = (float*)d_out;                         // B x 128

    const int* src = eidx;        // edge_index[0]
    const int* dst = eidx + E;    // edge_index[1]

    // carve workspace
    char* p = (char*)d_ws;
    auto take = [&](size_t bytes) { char* r = p; p += alignUp(bytes, 256); return r; };
    float*              deg      = (float*)take(sizeof(float) * N);     // deg -> dinv
    float*              hmat     = (float*)take(sizeof(float) * N * HID); // h, later new_x
    float*              hagg     = (float*)take(sizeof(float) * N * HID); // agg -> hrelu
    float*              escore   = (float*)take(sizeof(float) * E);     // e -> ex -> score
    unsigned*           mseg     = (unsigned*)take(sizeof(unsigned) * N);
    float*              den      = (float*)take(sizeof(float) * N);
    unsigned long long* keys     = (unsigned long long*)take(sizeof(unsigned long long) * SORT_N);
    int*                remaining= (int*)take(sizeof(int) * N);
    int*                cluster  = (int*)take(sizeof(int) * N);
    float*              scbuf    = (float*)take(sizeof(float) * N);
    int*                nbatch   = (int*)take(sizeof(int) * N);
    int*                counters = (int*)take(sizeof(int) * 8);
    float*              cntb     = (float*)take(sizeof(float) * B);
    float*              meanb    = (float*)take(sizeof(float) * B * HID);
    unsigned*           mxb      = (unsigned*)take(sizeof(unsigned) * B * HID);

    const int TB = 256;
    const int gN   = (N + TB - 1) / TB;
    const int gE   = (E + TB - 1) / TB;
    const long long nh = (long long)N * HID;
    const long long eh = (long long)E * HID;
    const int gNH  = (int)((nh + TB - 1) / TB);
    const int gEH  = (int)((eh + TB - 1) / TB);

    // ---- GCN conv ----
    k_init_nodes<<<gN, TB, 0, stream>>>(deg, remaining, cluster, scbuf, mseg, den, nbatch);
    k_deg_edges <<<gE, TB, 0, stream>>>(dst, deg);
    k_dinv      <<<gN, TB, 0, stream>>>(deg);
    k_gemm_wmma <<<N / 16, 128, 0, stream>>>(x, W, hmat);
    k_selfloop  <<<gNH, TB, 0, stream>>>(hmat, deg, hagg);
    k_edge_scatter<<<gEH, TB, 0, stream>>>(hmat, deg, src, dst, hagg);
    k_bias_relu <<<gNH, TB, 0, stream>>>(hagg, bias);

    // ---- edge pool scores ----
    k_edge_score<<<gE, TB, 0, stream>>>(hagg, src, dst, w_pool, b_pool, escore, mseg);
    k_ex_den    <<<gE, TB, 0, stream>>>(escore, dst, mseg, den);
    k_score     <<<gE, TB, 0, stream>>>(escore, dst, den);

    // ---- stable descending argsort (bitonic over packed 64-bit keys, L2-resident) ----
    k_init_keys<<<SORT_N / TB, TB, 0, stream>>>(escore, keys);
    for (int k = 2; k <= SORT_N; k <<= 1)
        for (int j = k >> 1; j > 0; j >>= 1)
            k_bitonic_step<<<SORT_N / TB, TB, 0, stream>>>(keys, j, k);

    // ---- sequential greedy matching ----
    k_greedy<<<1, 1, 0, stream>>>(keys, src, dst, remaining, cluster, scbuf, escore, counters);

    // ---- coarsened features (reuse hmat as new_x) ----
    k_clear_newx  <<<gNH, TB, 0, stream>>>(hmat);
    k_pool_scatter<<<gNH, TB, 0, stream>>>(hagg, cluster, batch, hmat, nbatch);
    k_scale_newx  <<<gNH, TB, 0, stream>>>(hmat, scbuf);

    // ---- global readout ----
    k_clear_readout  <<<(B * HID + TB - 1) / TB, TB, 0, stream>>>(meanb, mxb, cntb);
    k_readout_scatter<<<gNH, TB, 0, stream>>>(hmat, nbatch, counters, meanb, mxb, cntb);
    k_finalize       <<<(B * HID + TB - 1) / TB, TB, 0, stream>>>(meanb, mxb, cntb, out);
}